// QAModel_3590592660148
// MI455X (gfx1250) — compile-verified
//
#include <hip/hip_runtime.h>
#include <hip/hip_bf16.h>

// ---------------------------------------------------------------------------
// QA model forward (DrQA-style) for MI455X / gfx1250.
// All GEMM-like work uses V_WMMA_F32_16X16X4_F32 (full-precision f32 WMMA):
// the workload is ~27 GFLOP and latency bound (sequential LSTM), so we keep
// reference f32 precision instead of chasing f16/fp8 peak. The LSTM K-loop is
// split into a branch-free LDS phase (recurrent h) and global phase (x), and
// next-timestep inputs are prefetched (global_prefetch_b8) to overlap the
// critical recurrent chain with memory.
// ---------------------------------------------------------------------------

constexpr int EMBd = 300;
constexpr int HIDd = 100;
constexpr int FFd  = 100;
constexpr int MAXA = 30;
constexpr int TCc  = 400;
constexpr int TQq  = 30;
constexpr int Bb   = 32;
constexpr int SPN  = TCc * MAXA;              // 12000 spans per batch row
constexpr int PSW  = EMBd + 2 * HIDd + EMBd;  // 800  (p_star width)
constexpr int G4H  = 4 * HIDd;                // 400  (gate width)

typedef float v2f __attribute__((ext_vector_type(2)));
typedef float v8f __attribute__((ext_vector_type(8)));

__device__ __forceinline__ v8f wmma4(v2f a, v2f b, v8f c) {
  // D(16x16,f32) += A(16x4,f32) * B(4x16,f32)
  return __builtin_amdgcn_wmma_f32_16x16x4_f32(
      /*neg_a=*/false, a, /*neg_b=*/false, b,
      /*c_mod=*/(short)0, c, /*reuse_a=*/false, /*reuse_b=*/false);
}

__device__ __forceinline__ float sigmoidf_(float x) {
  return 1.0f / (1.0f + expf(-x));
}

// ---------------------------------------------------------------------------
// Embedding gather: out[t*B+b][e] = emb[tok[t*B+b]][e]
// ---------------------------------------------------------------------------
__global__ void gather_emb_kernel(const int* __restrict__ tok,
                                  const float* __restrict__ emb,
                                  float* __restrict__ out, int ntok) {
  int idx = blockIdx.x * blockDim.x + threadIdx.x;
  if (idx >= ntok * EMBd) return;
  int t = idx / EMBd, e = idx - t * EMBd;
  out[idx] = emb[(size_t)tok[t] * EMBd + e];
}

// ---------------------------------------------------------------------------
// Persistent LSTM over a whole sequence. blockIdx.x==0: forward (Wf/bf, out
// offset 0); blockIdx.x==1: reverse (Wr/br, out offset HIDd). Reverse writes
// its hidden state at the un-reversed time index (== h_rev[::-1] in the ref).
// Gate GEMM G(32x400) = [h|x](32x(HIDd+nin)) * W^T via f32 WMMA, K-loop split
// into a branch-free LDS (h) phase and global (x) phase.
// ---------------------------------------------------------------------------
__global__ __launch_bounds__(256) void lstm_seq_kernel(
    const float* __restrict__ x, const float* __restrict__ mask,
    const float* __restrict__ Wf, const float* __restrict__ bf,
    const float* __restrict__ Wr, const float* __restrict__ br,
    float* __restrict__ out, int T, int nin, int ostride) {
  const bool rev = (blockIdx.x == 1);
  const float* W    = rev ? Wr : Wf;
  const float* bias = rev ? br : bf;
  const int ooff    = rev ? HIDd : 0;
  const int K = HIDd + nin;            // 400 / 300 / 900 — all multiples of 4
  const int tid = threadIdx.x;
  const int lane = tid & 31;
  const int wid  = tid >> 5;
  const int nwaves = blockDim.x >> 5;

  __shared__ float sh_h[Bb * HIDd];    // 12.8 KB
  __shared__ float sh_c[Bb * HIDd];    // 12.8 KB
  __shared__ float sh_g[Bb * G4H];     // 51.2 KB

  for (int i = tid; i < Bb * HIDd; i += blockDim.x) { sh_h[i] = 0.f; sh_c[i] = 0.f; }
  __syncthreads();

  const int r15   = lane & 15;
  const int khalf = (lane >> 4) << 1;  // 0 for lanes 0-15, 2 for lanes 16-31
  const int mbase = (lane >> 4) << 3;  // C rows: vgpr r -> M = r + mbase

  for (int s = 0; s < T; ++s) {
    const int t = rev ? (T - 1 - s) : s;
    const float* xt = x + (size_t)t * Bb * nin;

    // Prefetch next timestep's input block (overlaps the recurrent chain).
    {
      const int tn = rev ? (t - 1) : (t + 1);
      if (tn >= 0 && tn < T) {
        const float* xn = x + (size_t)tn * Bb * nin;
        for (int i = tid * 16; i < Bb * nin; i += blockDim.x * 16)
          __builtin_prefetch(xn + i, 0, 3);   // global_prefetch_b8
      }
    }

    for (int nt = wid; nt < G4H / 16; nt += nwaves) {   // 25 N-tiles
      const int n0  = nt << 4;
      const int col = n0 + r15;                         // < 400 always
      const float* Wcol = W + (size_t)col * K;
      v8f acc0 = {}, acc1 = {};

      // Phase 1: recurrent h from LDS (k in [0, HIDd))
      for (int k0 = 0; k0 < HIDd; k0 += 4) {
        const int kk = k0 + khalf;
        v2f bfr; bfr.x = Wcol[kk]; bfr.y = Wcol[kk + 1];
        v2f a0, a1;
        a0.x = sh_h[r15 * HIDd + kk];        a0.y = sh_h[r15 * HIDd + kk + 1];
        a1.x = sh_h[(16 + r15) * HIDd + kk]; a1.y = sh_h[(16 + r15) * HIDd + kk + 1];
        acc0 = wmma4(a0, bfr, acc0);
        acc1 = wmma4(a1, bfr, acc1);
      }

      // Phase 2: input x from global (k in [HIDd, K))
      const float* x0 = xt + (size_t)r15 * nin;
      const float* x1 = xt + (size_t)(16 + r15) * nin;
      for (int k0 = HIDd; k0 < K; k0 += 4) {
        const int kk = k0 + khalf;
        v2f bfr; bfr.x = Wcol[kk]; bfr.y = Wcol[kk + 1];
        const int kx = kk - HIDd;
        v2f a0, a1;
        a0.x = x0[kx]; a0.y = x0[kx + 1];
        a1.x = x1[kx]; a1.y = x1[kx + 1];
        acc0 = wmma4(a0, bfr, acc0);
        acc1 = wmma4(a1, bfr, acc1);
      }

      const float bv = bias[col];
#pragma unroll
      for (int r2 = 0; r2 < 8; ++r2) {
        sh_g[(mbase + r2) * G4H + col]      = acc0[r2] + bv;
        sh_g[(16 + mbase + r2) * G4H + col] = acc1[r2] + bv;
      }
    }
    __syncthreads();

    for (int i = tid; i < Bb * HIDd; i += blockDim.x) {
      const int bq = i / HIDd, j = i - bq * HIDd;
      const float* g = sh_g + (size_t)bq * G4H;
      const float fg = sigmoidf_(g[j]);
      const float ig = sigmoidf_(g[HIDd + j]);
      const float og = sigmoidf_(g[2 * HIDd + j]);
      const float ct = tanhf(g[3 * HIDd + j]);
      const float co = sh_c[i];
      float cn = fg * co + ig * ct;
      const float m = mask[(size_t)t * Bb + bq];
      cn = m * cn + (1.f - m) * co;
      const float ho = sh_h[i];
      float hn = og * tanhf(cn);
      hn = m * hn + (1.f - m) * ho;
      sh_c[i] = cn;
      sh_h[i] = hn;
      out[((size_t)t * Bb + bq) * ostride + ooff + j] = hn;
    }
    __syncthreads();
  }
}

// ---------------------------------------------------------------------------
// Generic WMMA linear: C[M][N] = A[M][K] * W^T (+bias). W row-major [N][K].
// grid.x = M/32 (M always a multiple of 32 here). N guarded (N=100).
// ---------------------------------------------------------------------------
__global__ __launch_bounds__(256) void wmma_linear_kernel(
    const float* __restrict__ A, const float* __restrict__ W,
    const float* __restrict__ bias, float* __restrict__ C,
    int M, int K, int N) {
  const int tid = threadIdx.x, lane = tid & 31, wid = tid >> 5;
  const int nwaves = blockDim.x >> 5;
  const int m0 = blockIdx.x * 32;
  const int r15 = lane & 15;
  const int khalf = (lane >> 4) << 1;
  const int mbase = (lane >> 4) << 3;
  const int ntiles = (N + 15) >> 4;
  for (int nt = wid; nt < ntiles; nt += nwaves) {
    const int n0 = nt << 4;
    const int col = n0 + r15;
    v8f acc0 = {}, acc1 = {};
    for (int k0 = 0; k0 < K; k0 += 4) {
      const int kk = k0 + khalf;
      v2f bfr; bfr.x = 0.f; bfr.y = 0.f;
      if (col < N) {
        bfr.x = W[(size_t)col * K + kk];
        bfr.y = W[(size_t)col * K + kk + 1];
      }
      v2f a0, a1;
      const float* Ar0 = A + (size_t)(m0 + r15) * K + kk;
      const float* Ar1 = A + (size_t)(m0 + 16 + r15) * K + kk;
      a0.x = Ar0[0]; a0.y = Ar0[1];
      a1.x = Ar1[0]; a1.y = Ar1[1];
      acc0 = wmma4(a0, bfr, acc0);
      acc1 = wmma4(a1, bfr, acc1);
    }
    if (col < N) {
      const float bv = bias ? bias[col] : 0.f;
#pragma unroll
      for (int r2 = 0; r2 < 8; ++r2) {
        C[(size_t)(m0 + mbase + r2) * N + col]      = acc0[r2] + bv;
        C[(size_t)(m0 + 16 + mbase + r2) * N + col] = acc1[r2] + bv;
      }
    }
  }
}

// ---------------------------------------------------------------------------
// scores[b][p][q] = sum_f fp[p][b][f] * fq[q][b][f]   (WMMA, one block per b)
// ---------------------------------------------------------------------------
__global__ __launch_bounds__(256) void attn_scores_kernel(
    const float* __restrict__ fp, const float* __restrict__ fq,
    float* __restrict__ scores) {
  const int b = blockIdx.x;
  const int tid = threadIdx.x, lane = tid & 31, wid = tid >> 5;
  const int nwaves = blockDim.x >> 5;
  const int r15 = lane & 15;
  const int khalf = (lane >> 4) << 1;
  const int mbase = (lane >> 4) << 3;
  const int MT = TCc / 16;             // 25
  const int NT = (TQq + 15) / 16;      // 2
  for (int tile = wid; tile < MT * NT; tile += nwaves) {
    const int mt = tile / NT, nt = tile - mt * NT;
    const int m0 = mt << 4, n0 = nt << 4;
    const int col = n0 + r15;
    v8f acc = {};
    for (int k0 = 0; k0 < FFd; k0 += 4) {
      const int kk = k0 + khalf;
      v2f a;
      a.x = fp[((size_t)(m0 + r15) * Bb + b) * FFd + kk];
      a.y = fp[((size_t)(m0 + r15) * Bb + b) * FFd + kk + 1];
      v2f bb; bb.x = 0.f; bb.y = 0.f;
      if (col < TQq) {
        bb.x = fq[((size_t)col * Bb + b) * FFd + kk];
        bb.y = fq[((size_t)col * Bb + b) * FFd + kk + 1];
      }
      acc = wmma4(a, bb, acc);
    }
    if (col < TQq) {
#pragma unroll
      for (int r2 = 0; r2 < 8; ++r2)
        scores[((size_t)b * TCc + m0 + mbase + r2) * TQq + col] = acc[r2];
    }
  }
}

// ---------------------------------------------------------------------------
// q_align[p][b][e] = sum_q aw[b][p][q] * eq[q][b][e]; written directly into
// p_star[p][b][500 + e]. One block per b. K=30 padded to 32 (zero-guarded).
// ---------------------------------------------------------------------------
__global__ __launch_bounds__(256) void q_align_kernel(
    const float* __restrict__ aw, const float* __restrict__ eq,
    float* __restrict__ pstar) {
  const int b = blockIdx.x;
  const int tid = threadIdx.x, lane = tid & 31, wid = tid >> 5;
  const int nwaves = blockDim.x >> 5;
  const int r15 = lane & 15;
  const int khalf = (lane >> 4) << 1;
  const int mbase = (lane >> 4) << 3;
  const int MT = TCc / 16;             // 25
  const int NT = (EMBd + 15) / 16;     // 19
  for (int tile = wid; tile < MT * NT; tile += nwaves) {
    const int mt = tile / NT, nt = tile - mt * NT;
    const int m0 = mt << 4, n0 = nt << 4;
    const int col = n0 + r15;
    v8f acc = {};
    for (int k0 = 0; k0 < 32; k0 += 4) {
      const int kk = k0 + khalf;
      v2f a; a.x = 0.f; a.y = 0.f;
      if (kk < TQq)     a.x = aw[((size_t)b * TCc + m0 + r15) * TQq + kk];
      if (kk + 1 < TQq) a.y = aw[((size_t)b * TCc + m0 + r15) * TQq + kk + 1];
      v2f bb; bb.x = 0.f; bb.y = 0.f;
      if (col < EMBd) {
        if (kk < TQq)     bb.x = eq[((size_t)kk * Bb + b) * EMBd + col];
        if (kk + 1 < TQq) bb.y = eq[((size_t)(kk + 1) * Bb + b) * EMBd + col];
      }
      acc = wmma4(a, bb, acc);
    }
    if (col < EMBd) {
#pragma unroll
      for (int r2 = 0; r2 < 8; ++r2)
        pstar[((size_t)(m0 + mbase + r2) * Bb + b) * PSW + 500 + col] = acc[r2];
    }
  }
}

// ---------------------------------------------------------------------------
// out[r] = X[r][:] . w   (rows x cols, cols = FFd)
// ---------------------------------------------------------------------------
__global__ void rowdot_kernel(const float* __restrict__ X,
                              const float* __restrict__ w,
                              float* __restrict__ out, int rows, int cols) {
  int r = blockIdx.x * blockDim.x + threadIdx.x;
  if (r >= rows) return;
  float s = 0.f;
  for (int j = 0; j < cols; ++j) s += X[(size_t)r * cols + j] * w[j];
  out[r] = s;
}

// ---------------------------------------------------------------------------
// w = _softmax_cols_mask(s, qmask): softmax over t per batch column b.
// ---------------------------------------------------------------------------
__global__ void qweight_softmax_kernel(const float* __restrict__ s,
                                       const float* __restrict__ qmask,
                                       float* __restrict__ w) {
  int b = threadIdx.x;
  if (b >= Bb) return;
  float xv[TQq];
  float mn = 3.4e38f;
  for (int t = 0; t < TQq; ++t) {
    float x = s[t * Bb + b] * qmask[t * Bb + b];
    xv[t] = x; mn = fminf(mn, x);
  }
  float mx = -3.4e38f;
  for (int t = 0; t < TQq; ++t) {
    float x = (xv[t] - mn) * qmask[t * Bb + b];
    xv[t] = x; mx = fmaxf(mx, x);
  }
  float sum = 0.f;
  for (int t = 0; t < TQq; ++t) {
    float e = qmask[t * Bb + b] * expf(xv[t] - mx);
    xv[t] = e; sum += e;
  }
  for (int t = 0; t < TQq; ++t) w[t * Bb + b] = xv[t] / sum;
}

// ---------------------------------------------------------------------------
// q_indep[b][k] = sum_t w[t][b] * qh[t][b][k]
// ---------------------------------------------------------------------------
__global__ void q_indep_kernel(const float* __restrict__ w,
                               const float* __restrict__ qh,
                               float* __restrict__ qind) {
  int idx = blockIdx.x * blockDim.x + threadIdx.x;
  if (idx >= Bb * 2 * HIDd) return;
  int b = idx / (2 * HIDd), k = idx - b * 2 * HIDd;
  float s = 0.f;
  for (int t = 0; t < TQq; ++t)
    s += w[t * Bb + b] * qh[((size_t)t * Bb + b) * (2 * HIDd) + k];
  qind[idx] = s;
}

// ---------------------------------------------------------------------------
// _softmax_depths_mask over q axis, in place on scores[b][p][:].
// ---------------------------------------------------------------------------
__global__ void aw_softmax_kernel(float* __restrict__ scores,
                                  const float* __restrict__ cmask,
                                  const float* __restrict__ qmask) {
  int r = blockIdx.x * blockDim.x + threadIdx.x;
  if (r >= Bb * TCc) return;
  int b = r / TCc, p = r - b * TCc;
  float mp = cmask[(size_t)p * Bb + b];
  float xv[TQq], pqm[TQq];
  float mn = 3.4e38f;
  for (int q = 0; q < TQq; ++q) {
    float pq = mp * qmask[q * Bb + b];
    pqm[q] = pq;
    float x = scores[(size_t)r * TQq + q] * pq;
    xv[q] = x; mn = fminf(mn, x);
  }
  float mx = -3.4e38f;
  for (int q = 0; q < TQq; ++q) {
    float x = (xv[q] - mn) * pqm[q];
    xv[q] = x; mx = fmaxf(mx, x);
  }
  float sum = 0.f;
  for (int q = 0; q < TQq; ++q) {
    float e = pqm[q] * expf(xv[q] - mx);
    xv[q] = e; sum += e;
  }
  float denom = sum + (sum == 0.f ? 1.f : 0.f);
  for (int q = 0; q < TQq; ++q) scores[(size_t)r * TQq + q] = xv[q] / denom;
}

// ---------------------------------------------------------------------------
// p_star[:, :, 0:300] = ec; p_star[:, :, 300:500] = q_indep (broadcast over p)
// ---------------------------------------------------------------------------
__global__ void pstar_fill_kernel(const float* __restrict__ ec,
                                  const float* __restrict__ qind,
                                  float* __restrict__ pstar) {
  int idx = blockIdx.x * blockDim.x + threadIdx.x;
  if (idx >= TCc * Bb * 500) return;
  int pb = idx / 500, e = idx - pb * 500;
  float v = (e < EMBd) ? ec[(size_t)pb * EMBd + e]
                       : qind[(size_t)(pb % Bb) * (2 * HIDd) + (e - EMBd)];
  pstar[(size_t)pb * PSW + e] = v;
}

// ---------------------------------------------------------------------------
// x[b][s] = (sum_f relu(stt[t][b][f] + end[t+a][b][f]) * Wsp[f]) * span_mask
// with s = t*MAXA + a, end zero-padded beyond TCc.
// ---------------------------------------------------------------------------
__global__ void span_score_kernel(const float* __restrict__ stt,
                                  const float* __restrict__ endl,
                                  const float* __restrict__ Wsp,
                                  const int* __restrict__ clens,
                                  float* __restrict__ xspan) {
  int idx = blockIdx.x * blockDim.x + threadIdx.x;
  if (idx >= Bb * SPN) return;
  int b = idx / SPN, s2 = idx - b * SPN;
  int t = s2 / MAXA, a = s2 - t * MAXA, te = t + a;
  const float* sp = stt + ((size_t)t * Bb + b) * FFd;
  float acc = 0.f;
  for (int f = 0; f < FFd; ++f) {
    float ev = 0.f;
    if (te < TCc) ev = endl[((size_t)te * Bb + b) * FFd + f];
    float v = sp[f] + ev;
    v = fmaxf(v, 0.f);
    acc += v * Wsp[f];
  }
  float msk = (te < clens[b]) ? 1.f : 0.f;
  xspan[idx] = acc * msk;
}

// ---------------------------------------------------------------------------
// Per-column (over batch) min subtraction + re-mask: x = (x - min_b x) * mask
// ---------------------------------------------------------------------------
__global__ void colmin_sub_kernel(float* __restrict__ xspan,
                                  const int* __restrict__ clens) {
  int s2 = blockIdx.x * blockDim.x + threadIdx.x;
  if (s2 >= SPN) return;
  int te = s2 / MAXA + s2 % MAXA;
  float mn = xspan[s2];
  for (int b = 1; b < Bb; ++b) mn = fminf(mn, xspan[(size_t)b * SPN + s2]);
  for (int b = 0; b < Bb; ++b) {
    float msk = (te < clens[b]) ? 1.f : 0.f;
    xspan[(size_t)b * SPN + s2] = (xspan[(size_t)b * SPN + s2] - mn) * msk;
  }
}

// ---------------------------------------------------------------------------
// Per-batch argmax (first occurrence) + masked log-sum-exp + xent.
// ---------------------------------------------------------------------------
__global__ __launch_bounds__(256) void perb_kernel(
    const float* __restrict__ xspan, const int* __restrict__ clens,
    const int* __restrict__ anss, float* __restrict__ xent,
    float* __restrict__ accb) {
  const int b = blockIdx.x;
  const int tid = threadIdx.x;
  __shared__ float rmax[256];
  __shared__ int   ridx[256];
  __shared__ float rsum[256];
  float best = -3.4e38f; int bi = SPN;
  for (int s = tid; s < SPN; s += 256) {
    float v = xspan[(size_t)b * SPN + s];
    if (v > best) { best = v; bi = s; }
  }
  rmax[tid] = best; ridx[tid] = bi;
  __syncthreads();
  for (int off = 128; off > 0; off >>= 1) {
    if (tid < off) {
      float vo = rmax[tid + off]; int io = ridx[tid + off];
      if (vo > rmax[tid] || (vo == rmax[tid] && io < ridx[tid])) {
        rmax[tid] = vo; ridx[tid] = io;
      }
    }
    __syncthreads();
  }
  const float mx = rmax[0];
  const int amax = ridx[0];
  float sum = 0.f;
  for (int s = tid; s < SPN; s += 256) {
    int te = s / MAXA + s % MAXA;
    if (te < clens[b]) sum += expf(xspan[(size_t)b * SPN + s] - mx);
  }
  rsum[tid] = sum;
  __syncthreads();
  for (int off = 128; off > 0; off >>= 1) {
    if (tid < off) rsum[tid] += rsum[tid + off];
    __syncthreads();
  }
  if (tid == 0) {
    float lse = logf(rsum[0]);
    int ans = anss[b];
    int te = ans / MAXA + ans % MAXA;
    float msk = (te < clens[b]) ? 1.f : 0.f;
    float xstar = (xspan[(size_t)b * SPN + ans] - mx) * msk;
    xent[b] = lse - xstar;
    accb[b] = (amax == ans) ? 1.f : 0.f;
  }
}

__global__ void final_kernel(const float* __restrict__ xent,
                             const float* __restrict__ accb,
                             float* __restrict__ out) {
  if (threadIdx.x == 0) {
    float L = 0.f, A = 0.f;
    for (int i = 0; i < Bb; ++i) { L += xent[i]; A += accb[i]; }
    out[0] = L / (float)Bb;   // loss
    out[1] = A / (float)Bb;   // acc
    out[2] = A;               // accs.sum()
    out[3] = L / (float)Bb;   // loss.sum() (scalar)
  }
}

// ---------------------------------------------------------------------------
extern "C" void kernel_launch(void* const* d_in, const int* in_sizes, int n_in,
                              void* d_out, int out_size, void* d_ws,
                              size_t ws_size, hipStream_t stream) {
  (void)in_sizes; (void)n_in; (void)out_size; (void)ws_size;
  const int*   contexts = (const int*)d_in[0];
  const float* cmask    = (const float*)d_in[1];
  const int*   questions= (const int*)d_in[2];
  const float* qmask    = (const float*)d_in[3];
  const int*   anss     = (const int*)d_in[4];
  const int*   clens    = (const int*)d_in[5];
  const float* word_emb = (const float*)d_in[6];
  const float* qW1f = (const float*)d_in[7],  *qb1f = (const float*)d_in[8];
  const float* qW1r = (const float*)d_in[9],  *qb1r = (const float*)d_in[10];
  const float* qW2f = (const float*)d_in[11], *qb2f = (const float*)d_in[12];
  const float* qW2r = (const float*)d_in[13], *qb2r = (const float*)d_in[14];
  const float* pW1f = (const float*)d_in[15], *pb1f = (const float*)d_in[16];
  const float* pW1r = (const float*)d_in[17], *pb1r = (const float*)d_in[18];
  const float* pW2f = (const float*)d_in[19], *pb2f = (const float*)d_in[20];
  const float* pW2r = (const float*)d_in[21], *pb2r = (const float*)d_in[22];
  const float* W1  = (const float*)d_in[23], *b1  = (const float*)d_in[24];
  const float* Ww  = (const float*)d_in[25];
  const float* Wqa = (const float*)d_in[26];
  const float* Was = (const float*)d_in[27], *bas = (const float*)d_in[28];
  const float* Wae = (const float*)d_in[29];
  const float* Wsp = (const float*)d_in[30];

  float* ws = (float*)d_ws;
  size_t off = 0;
  auto alloc = [&](size_t n) { float* p = ws + off; off += n; return p; };

  float* eq     = alloc((size_t)TQq * Bb * EMBd);       // 288000
  float* ec     = alloc((size_t)TCc * Bb * EMBd);       // 3.84M
  float* inp2q  = alloc((size_t)TQq * Bb * 2 * HIDd);
  float* qh     = alloc((size_t)TQq * Bb * 2 * HIDd);
  float* tmpW1  = alloc((size_t)TQq * Bb * FFd);
  float* sbuf   = alloc((size_t)TQq * Bb);
  float* wbuf   = alloc((size_t)TQq * Bb);
  float* qind   = alloc((size_t)Bb * 2 * HIDd);
  float* fq     = alloc((size_t)TQq * Bb * FFd);
  float* fp     = alloc((size_t)TCc * Bb * FFd);
  float* scores = alloc((size_t)Bb * TCc * TQq);
  float* pstar  = alloc((size_t)TCc * Bb * PSW);        // 10.24M
  float* inp2p  = alloc((size_t)TCc * Bb * 2 * HIDd);
  float* ph     = alloc((size_t)TCc * Bb * 2 * HIDd);
  float* stt    = alloc((size_t)TCc * Bb * FFd);
  float* endb   = alloc((size_t)TCc * Bb * FFd);
  float* xspan  = alloc((size_t)Bb * SPN);
  float* xent   = alloc((size_t)Bb);
  float* accb   = alloc((size_t)Bb);

  // 1. Embedding gathers
  gather_emb_kernel<<<(TQq * Bb * EMBd + 255) / 256, 256, 0, stream>>>(
      questions, word_emb, eq, TQq * Bb);
  gather_emb_kernel<<<(TCc * Bb * EMBd + 255) / 256, 256, 0, stream>>>(
      contexts, word_emb, ec, TCc * Bb);

  // 2. Question encoder (bi-LSTM x2)
  lstm_seq_kernel<<<2, 256, 0, stream>>>(eq, qmask, qW1f, qb1f, qW1r, qb1r,
                                         inp2q, TQq, EMBd, 2 * HIDd);
  lstm_seq_kernel<<<2, 256, 0, stream>>>(inp2q, qmask, qW2f, qb2f, qW2r, qb2r,
                                         qh, TQq, 2 * HIDd, 2 * HIDd);

  // 3. Question-independent attention weights
  wmma_linear_kernel<<<(TQq * Bb) / 32, 256, 0, stream>>>(qh, W1, b1, tmpW1,
                                                          TQq * Bb, 2 * HIDd, FFd);
  rowdot_kernel<<<(TQq * Bb + 255) / 256, 256, 0, stream>>>(tmpW1, Ww, sbuf,
                                                            TQq * Bb, FFd);
  qweight_softmax_kernel<<<1, 32, 0, stream>>>(sbuf, qmask, wbuf);
  q_indep_kernel<<<(Bb * 2 * HIDd + 255) / 256, 256, 0, stream>>>(wbuf, qh, qind);

  // 4. Aligned question attention
  wmma_linear_kernel<<<(TQq * Bb) / 32, 256, 0, stream>>>(eq, Wqa, nullptr, fq,
                                                          TQq * Bb, EMBd, FFd);
  wmma_linear_kernel<<<(TCc * Bb) / 32, 256, 0, stream>>>(ec, Wqa, nullptr, fp,
                                                          TCc * Bb, EMBd, FFd);
  attn_scores_kernel<<<Bb, 256, 0, stream>>>(fp, fq, scores);
  aw_softmax_kernel<<<(Bb * TCc + 255) / 256, 256, 0, stream>>>(scores, cmask, qmask);

  // 5. p_star = [ec | q_indep | q_align]
  pstar_fill_kernel<<<(TCc * Bb * 500 + 255) / 256, 256, 0, stream>>>(ec, qind, pstar);
  q_align_kernel<<<Bb, 256, 0, stream>>>(scores, eq, pstar);

  // 6. Passage encoder (bi-LSTM x2)
  lstm_seq_kernel<<<2, 256, 0, stream>>>(pstar, cmask, pW1f, pb1f, pW1r, pb1r,
                                         inp2p, TCc, PSW, 2 * HIDd);
  lstm_seq_kernel<<<2, 256, 0, stream>>>(inp2p, cmask, pW2f, pb2f, pW2r, pb2r,
                                         ph, TCc, 2 * HIDd, 2 * HIDd);

  // 7. Span start/end projections
  wmma_linear_kernel<<<(TCc * Bb) / 32, 256, 0, stream>>>(ph, Was, bas, stt,
                                                          TCc * Bb, 2 * HIDd, FFd);
  wmma_linear_kernel<<<(TCc * Bb) / 32, 256, 0, stream>>>(ph, Wae, nullptr, endb,
                                                          TCc * Bb, 2 * HIDd, FFd);

  // 8. Span scoring, normalization, loss/acc
  span_score_kernel<<<(Bb * SPN + 255) / 256, 256, 0, stream>>>(stt, endb, Wsp,
                                                                clens, xspan);
  colmin_sub_kernel<<<(SPN + 255) / 256, 256, 0, stream>>>(xspan, clens);
  perb_kernel<<<Bb, 256, 0, stream>>>(xspan, clens, anss, xent, accb);
  final_kernel<<<1, 32, 0, stream>>>(xent, accb, (float*)d_out);
}